// PoseDisentangler_32014686224789
// MI455X (gfx1250) — compile-verified
//
#include <hip/hip_runtime.h>
#include <stdint.h>

#define JOINTS          17
#define JOINT_FLOATS    (JOINTS * 3)          // 51 floats = 204 B per pose
#define POSES_PER_BLOCK 64
#define THREADS         256                    // 8 wave32s
#define TILE_DWORDS     (POSES_PER_BLOCK * JOINT_FLOATS)  // 3264 dwords = 13056 B
#define TILE_B128       (TILE_DWORDS / 4)                 // 816 16B chunks
#define TAIL_LANES      (TILE_B128 - 3 * THREADS)         // 48

// H36M parents for bones 0..15 packed as 4-bit nibbles (bone j -> nibble j):
// [0,1,2,0,4,5,0,7,8,9,8,11,12,8,14,15]
#define PARENT_NIBBLES 0xFE8CB89870540210ULL

typedef float v4f __attribute__((ext_vector_type(4)));   // native vector for NT stores

__global__ __launch_bounds__(THREADS)
void pose_bones_kernel(const float* __restrict__ in,
                       float* __restrict__ out_len,
                       float* __restrict__ out_dir)
{
    __shared__ float smem[TILE_DWORDS];
    const int tid = threadIdx.x;

    // ---- CDNA5 async bulk copy: global -> LDS (ASYNCcnt-tracked) ----
    // IOFFSET is added to BOTH the LDS and the global address, so one
    // (lds,addr) pair feeds all four issues via immediate offsets.
    const float* gbase = in + (size_t)blockIdx.x * TILE_DWORDS;
    const uint32_t lds0 = (uint32_t)(uintptr_t)(&smem[0]);   // low 32 bits = LDS byte addr
    const uint32_t loff = lds0 + (uint32_t)tid * 16u;
    const float*   g    = gbase + tid * 4;                   // 16 contiguous bytes / lane

    asm volatile("global_load_async_to_lds_b128 %0, %1, off th:TH_LOAD_NT"
                 :: "v"(loff), "v"(g) : "memory");
    asm volatile("global_load_async_to_lds_b128 %0, %1, off offset:4096 th:TH_LOAD_NT"
                 :: "v"(loff), "v"(g) : "memory");
    asm volatile("global_load_async_to_lds_b128 %0, %1, off offset:8192 th:TH_LOAD_NT"
                 :: "v"(loff), "v"(g) : "memory");
    if (tid < TAIL_LANES) {
        asm volatile("global_load_async_to_lds_b128 %0, %1, off offset:12288 th:TH_LOAD_NT"
                     :: "v"(loff), "v"(g) : "memory");
    }

#if __has_builtin(__builtin_amdgcn_s_wait_asynccnt)
    __builtin_amdgcn_s_wait_asynccnt(0);
#else
    asm volatile("s_wait_asynccnt 0" ::: "memory");
#endif
    __syncthreads();

    // ---- compute: 4 threads per pose, 4 bones per thread ----
    const int sub    = tid & 3;
    const int poseL  = tid >> 2;
    const size_t pg  = (size_t)blockIdx.x * POSES_PER_BLOCK + poseL;
    const float* P   = &smem[poseL * JOINT_FLOATS];

    float len[4], dir[12];
    #pragma unroll
    for (int k = 0; k < 4; ++k) {
        int j = sub * 4 + k;                                     // bone 0..15
        int c = 3 * (j + 1);                                     // child joint
        int p = 3 * (int)((PARENT_NIBBLES >> (4 * j)) & 0xFull); // parent joint
        float vx = P[c + 0] - P[p + 0];
        float vy = P[c + 1] - P[p + 1];
        float vz = P[c + 2] - P[p + 2];
        float l  = sqrtf(fmaf(vx, vx, fmaf(vy, vy, vz * vz)));
        float iv = 1.0f / fmaxf(l, 1e-12f);
        len[k] = l;
        dir[k * 3 + 0] = vx * iv;
        dir[k * 3 + 1] = vy * iv;
        dir[k * 3 + 2] = vz * iv;
    }

    // ---- streamed outputs: 16B-aligned non-temporal vector stores ----
    v4f lv = { len[0], len[1], len[2], len[3] };
    __builtin_nontemporal_store(lv, (v4f*)(out_len + pg * 16 + sub * 4));

    v4f* dp = (v4f*)(out_dir + pg * 48 + sub * 12);
    v4f d0 = { dir[0], dir[1], dir[2],  dir[3]  };
    v4f d1 = { dir[4], dir[5], dir[6],  dir[7]  };
    v4f d2 = { dir[8], dir[9], dir[10], dir[11] };
    __builtin_nontemporal_store(d0, dp + 0);
    __builtin_nontemporal_store(d1, dp + 1);
    __builtin_nontemporal_store(d2, dp + 2);
}

extern "C" void kernel_launch(void* const* d_in, const int* in_sizes, int n_in,
                              void* d_out, int out_size, void* d_ws, size_t ws_size,
                              hipStream_t stream) {
    (void)n_in; (void)out_size; (void)d_ws; (void)ws_size;
    const float* in = (const float*)d_in[0];
    float* out = (float*)d_out;

    const size_t nposes = (size_t)in_sizes[0] / JOINT_FLOATS;   // 524288
    float* out_len = out;                      // P*16 floats
    float* out_dir = out + nposes * 16;        // P*48 floats

    dim3 grid((unsigned)(nposes / POSES_PER_BLOCK));            // 8192 blocks
    pose_bones_kernel<<<grid, THREADS, 0, stream>>>(in, out_len, out_dir);
}